// NlsConv2d_31224412242793
// MI455X (gfx1250) — compile-verified
//
#include <hip/hip_runtime.h>
#include <math.h>

typedef __attribute__((ext_vector_type(2))) float v2f;
typedef __attribute__((ext_vector_type(8))) float v8f;

#define BATCH 2
#define CCH   32
#define OCH   32
#define HH    128
#define WW    128
#define NPIX  (HH*WW)      // 16384
#define CIN   66
#define WSZ   7
#define WIN   49
#define TOPK  9
#define KK    (CCH*9)      // 288

// ---------------- Kernel 1: the two small 3x3 convs (offset + mask) ----------------
__global__ __launch_bounds__(256) void k_convs(
    const float* __restrict__ fi, const float* __restrict__ fj,
    const float* __restrict__ flow,
    const float* __restrict__ w_off, const float* __restrict__ b_off,
    const float* __restrict__ w_mod, const float* __restrict__ b_mod,
    float* __restrict__ soff, float* __restrict__ maskb)
{
    int t = blockIdx.x * blockDim.x + threadIdx.x;
    if (t >= BATCH * NPIX) return;
    int b = t / NPIX;
    int n = t % NPIX;
    int h = n / WW, w = n % WW;

    float acc[11];
    acc[0] = b_off[0];
    acc[1] = b_off[1];
    #pragma unroll
    for (int m = 0; m < 9; ++m) acc[2 + m] = b_mod[m];

    const float* fib = fi   + (size_t)b * CCH * NPIX;
    const float* fjb = fj   + (size_t)b * CCH * NPIX;
    const float* flb = flow + (size_t)b * 2   * NPIX;

    for (int ty = -1; ty <= 1; ++ty) {
        int yy = h + ty;
        if (yy < 0 || yy >= HH) continue;
        for (int tx = -1; tx <= 1; ++tx) {
            int xx = w + tx;
            if (xx < 0 || xx >= WW) continue;
            int tap = (ty + 1) * 3 + (tx + 1);
            int pix = yy * WW + xx;
            for (int ic = 0; ic < CIN; ++ic) {
                float v;
                if (ic < CCH)          v = fib[ic * NPIX + pix];
                else if (ic < 2*CCH)   v = fjb[(ic - CCH) * NPIX + pix];
                else                   v = flb[(ic - 2*CCH) * NPIX + pix];
                acc[0] += w_off[(0 * CIN + ic) * 9 + tap] * v;
                acc[1] += w_off[(1 * CIN + ic) * 9 + tap] * v;
                #pragma unroll
                for (int m = 0; m < 9; ++m)
                    acc[2 + m] += w_mod[(m * CIN + ic) * 9 + tap] * v;
            }
        }
    }
    soff[((size_t)b * 2 + 0) * NPIX + n] = acc[0];
    soff[((size_t)b * 2 + 1) * NPIX + n] = acc[1];
    #pragma unroll
    for (int m = 0; m < 9; ++m)
        maskb[((size_t)b * 9 + m) * NPIX + n] = 2.0f / (1.0f + expf(-acc[2 + m]));
}

// ------- Kernel 2: correlation over 7x7 window, top-9, masked resample -> Act -------
__global__ __launch_bounds__(256) void k_sample(
    const float* __restrict__ fi, const float* __restrict__ fj,
    const float* __restrict__ soff, const float* __restrict__ maskb,
    float* __restrict__ act)
{
    int t = blockIdx.x * blockDim.x + threadIdx.x;
    if (t >= BATCH * NPIX) return;
    int b = t / NPIX, n = t % NPIX;
    int h = n / WW, w = n % WW;

    const float* fjb = fj + (size_t)b * CCH * NPIX;
    const float* fib = fi + (size_t)b * CCH * NPIX;

    float fvec[CCH];
    #pragma unroll
    for (int c = 0; c < CCH; ++c) fvec[c] = fib[c * NPIX + n];

    float soy = soff[((size_t)b * 2 + 0) * NPIX + n];
    float sox = soff[((size_t)b * 2 + 1) * NPIX + n];

    // correlation scores over the 7x7 window (clip-padded bilinear)
    float dist[WIN];
    for (int pp = 0; pp < WIN; ++pp) {
        float dy = (float)(pp / WSZ) - 3.0f;
        float dx = (float)(pp % WSZ) - 3.0f;
        float cy = fminf(fmaxf((float)h + soy + dy, 0.0f), (float)(HH - 1));
        float cx = fminf(fmaxf((float)w + sox + dx, 0.0f), (float)(WW - 1));
        float y0 = floorf(cy), x0 = floorf(cx);
        float wy1 = cy - y0, wx1 = cx - x0;
        int iy0 = (int)y0, ix0 = (int)x0;
        int iy1 = min(iy0 + 1, HH - 1), ix1 = min(ix0 + 1, WW - 1);
        float w00 = (1.0f - wy1) * (1.0f - wx1);
        float w01 = (1.0f - wy1) * wx1;
        float w10 = wy1 * (1.0f - wx1);
        float w11 = wy1 * wx1;
        int i00 = iy0 * WW + ix0, i01 = iy0 * WW + ix1;
        int i10 = iy1 * WW + ix0, i11 = iy1 * WW + ix1;
        float d = 0.0f;
        for (int c = 0; c < CCH; ++c) {
            const float* pc = fjb + c * NPIX;
            float s = w00 * pc[i00] + w01 * pc[i01] + w10 * pc[i10] + w11 * pc[i11];
            d += fvec[c] * s;
        }
        dist[pp] = d;
    }

    // top-9, descending, first index wins ties (matches lax.top_k)
    unsigned long long used = 0ull;
    int sel[TOPK];
    for (int k = 0; k < TOPK; ++k) {
        float best = -INFINITY; int bi = 0;
        for (int pp = 0; pp < WIN; ++pp) {
            if ((used >> pp) & 1ull) continue;
            if (dist[pp] > best) { best = dist[pp]; bi = pp; }
        }
        sel[k] = bi; used |= (1ull << bi);
    }

    float* actp = act + ((size_t)b * NPIX + n) * KK;
    for (int k = 0; k < TOPK; ++k) {
        int pp = sel[k];
        float dy = (float)(pp / WSZ) - 3.0f;
        float dx = (float)(pp % WSZ) - 3.0f;
        float cy = fminf(fmaxf((float)h + soy + dy, 0.0f), (float)(HH - 1));
        float cx = fminf(fmaxf((float)w + sox + dx, 0.0f), (float)(WW - 1));
        float offy = cy - (float)h;
        float offx = cx - (float)w;
        float kyf = (float)(k / 3) - 1.0f;
        float kxf = (float)(k % 3) - 1.0f;
        // NOTE: reference's 0::2 / 1::2 slicing swaps the offsets (oy<-off_x, ox<-off_y)
        float sy = (float)h + kyf + offx;
        float sx = (float)w + kxf + offy;

        float y0 = floorf(sy), x0 = floorf(sx);
        float wy1 = sy - y0, wx1 = sx - x0;
        int iy0 = (int)y0, ix0 = (int)x0;
        int iy1 = iy0 + 1, ix1 = ix0 + 1;
        bool vy0 = (iy0 >= 0) && (iy0 <= HH - 1);
        bool vy1 = (iy1 >= 0) && (iy1 <= HH - 1);
        bool vx0 = (ix0 >= 0) && (ix0 <= WW - 1);
        bool vx1 = (ix1 >= 0) && (ix1 <= WW - 1);
        float w00 = (vy0 && vx0) ? (1.0f - wy1) * (1.0f - wx1) : 0.0f;
        float w01 = (vy0 && vx1) ? (1.0f - wy1) * wx1 : 0.0f;
        float w10 = (vy1 && vx0) ? wy1 * (1.0f - wx1) : 0.0f;
        float w11 = (vy1 && vx1) ? wy1 * wx1 : 0.0f;
        int cy0 = min(max(iy0, 0), HH - 1), cy1 = min(max(iy1, 0), HH - 1);
        int cx0 = min(max(ix0, 0), WW - 1), cx1 = min(max(ix1, 0), WW - 1);
        int i00 = cy0 * WW + cx0, i01 = cy0 * WW + cx1;
        int i10 = cy1 * WW + cx0, i11 = cy1 * WW + cx1;

        float mk = maskb[((size_t)b * 9 + k) * NPIX + n];
        for (int c = 0; c < CCH; ++c) {
            const float* pc = fjb + c * NPIX;
            float s = w00 * pc[i00] + w01 * pc[i01] + w10 * pc[i10] + w11 * pc[i11];
            actp[c * 9 + k] = s * mk;
        }
    }
}

// ---- Kernel 3: out[b,o,n] = sum_k w_reg[o,k] * Act[b,n,k] + b_reg[o], via f32 WMMA ----
// One wave per 16x16 tile. A = w_reg (M=o rows), B = Act (N=pixel cols), K = 288.
// f32 WMMA 16x16x4 fragment layout (ISA 7.12.2): A lane<16 holds M=lane,K={0,1};
// lane>=16 holds M=lane-16,K={2,3}. B assumed symmetric: lane<16 N=lane,K={0,1}; etc.
__global__ __launch_bounds__(256) void k_gemm_wmma(
    const float* __restrict__ act, const float* __restrict__ w_reg,
    const float* __restrict__ b_reg, float* __restrict__ out)
{
    int wid  = (int)((blockIdx.x * blockDim.x + threadIdx.x) >> 5);
    int lane = threadIdx.x & 31;
    int nt = wid & 1023;          // 1024 N-tiles per (b, m-tile)
    int mt = (wid >> 10) & 1;     // 2 M-tiles (O = 32)
    int b  = wid >> 11;           // batch
    int n0 = nt * 16, m0 = mt * 16;
    int lm = lane & 15;
    int hi = lane >> 4;

    const float* Ap = w_reg + (size_t)(m0 + lm) * KK + 2 * hi;
    const float* Bp = act   + ((size_t)b * NPIX + n0 + lm) * KK + 2 * hi;

    v8f acc = {};
    #pragma unroll 4
    for (int k = 0; k < KK; k += 4) {
        v2f a  = *(const v2f*)(Ap + k);
        v2f bb = *(const v2f*)(Bp + k);
        acc = __builtin_amdgcn_wmma_f32_16x16x4_f32(
            /*neg_a=*/false, a, /*neg_b=*/false, bb,
            /*c_mod=*/(short)0, acc, /*reuse_a=*/false, /*reuse_b=*/false);
    }
    #pragma unroll
    for (int r = 0; r < 8; ++r) {
        int o = m0 + r + 8 * hi;  // C/D layout: VGPR r -> rows M=r (lanes<16) / M=r+8
        out[((size_t)b * OCH + o) * NPIX + n0 + lm] = acc[r] + b_reg[o];
    }
}

extern "C" void kernel_launch(void* const* d_in, const int* in_sizes, int n_in,
                              void* d_out, int out_size, void* d_ws, size_t ws_size,
                              hipStream_t stream) {
    (void)in_sizes; (void)n_in; (void)out_size; (void)ws_size;
    const float* fi    = (const float*)d_in[0];
    const float* fj    = (const float*)d_in[1];
    const float* flow  = (const float*)d_in[2];
    const float* w_off = (const float*)d_in[3];
    const float* b_off = (const float*)d_in[4];
    const float* w_mod = (const float*)d_in[5];
    const float* b_mod = (const float*)d_in[6];
    const float* w_reg = (const float*)d_in[7];
    const float* b_reg = (const float*)d_in[8];
    float* out = (float*)d_out;

    // workspace layout: Act[B][N][288] | soff[B][2][N] | mask[B][9][N]  (~39.2 MB)
    float* act   = (float*)d_ws;
    float* soff  = act  + (size_t)BATCH * NPIX * KK;
    float* maskb = soff + (size_t)BATCH * 2 * NPIX;

    k_convs<<<(BATCH * NPIX) / 256, 256, 0, stream>>>(
        fi, fj, flow, w_off, b_off, w_mod, b_mod, soff, maskb);
    k_sample<<<(BATCH * NPIX) / 256, 256, 0, stream>>>(
        fi, fj, soff, maskb, act);
    // 4096 tiles -> 4096 waves -> 131072 threads -> 512 blocks of 256
    k_gemm_wmma<<<(BATCH * 2 * 1024 * 32) / 256, 256, 0, stream>>>(
        act, w_reg, b_reg, out);
}